// GCN_59571196395871
// MI455X (gfx1250) — compile-verified
//
#include <hip/hip_runtime.h>
#include <hip/hip_bf16.h>
#include <math.h>

#define N_NODES 100000
#define N_REL   1000
#define N_EDGES 1600000
#define DIM     128
#define BATCH   8192

typedef __attribute__((ext_vector_type(2))) float v2f;
typedef __attribute__((ext_vector_type(8))) float v8f;

// ---------------------------------------------------------------------------
// Degree / normalization: deg[i] = 1 (self loop) + count(row == i); dis = deg^-0.5
// ---------------------------------------------------------------------------
__global__ void init_deg_kernel(float* __restrict__ deg) {
    int i = blockIdx.x * blockDim.x + threadIdx.x;
    if (i < N_NODES) deg[i] = 1.0f;   // self loop contributes 1
}

__global__ void deg_accum_kernel(const int* __restrict__ row, float* __restrict__ deg) {
    int e = blockIdx.x * blockDim.x + threadIdx.x;
    if (e < N_EDGES) atomicAdd(&deg[row[e]], 1.0f);
}

__global__ void deg_rsqrt_kernel(float* __restrict__ deg) {
    int i = blockIdx.x * blockDim.x + threadIdx.x;
    if (i < N_NODES) deg[i] = rsqrtf(deg[i]);
}

// ---------------------------------------------------------------------------
// H = act(X) @ W + b  via V_WMMA_F32_16X16X4_F32 (exact f32 path).
// One wave per 16x16 output tile; block = 8 waves = the 8 N-tiles of one
// 16-row strip. 32 WMMAs per tile (K = 128 in steps of 4).
//
// Operand layouts (ISA 7.12.2, wave32):
//   A (16x4):  lane L holds row M = L%16;  VGPR0/1 = K = (L/16)*2 + {0,1}
//   B (4x16):  lane L holds col N = L%16;  VGPR0/1 = K = (L/16)*2 + {0,1}
//   C/D:       lane L, VGPR r -> M = r + (L/16)*8, N = L%16
// ---------------------------------------------------------------------------
template <bool RELU_IN>
__global__ void gemm_wmma_kernel(const float* __restrict__ X,
                                 const float* __restrict__ W,
                                 const float* __restrict__ bias,
                                 float* __restrict__ H) {
    const int wave = threadIdx.x >> 5;   // 0..7 -> N tile
    const int lane = threadIdx.x & 31;
    const int half = lane >> 4;          // selects K pair / M half
    const int l16  = lane & 15;
    const int m0   = blockIdx.x << 4;
    const int n    = (wave << 4) + l16;

    // Accumulator initialized with bias (broadcast along M)
    v8f acc;
    const float bv = bias[n];
#pragma unroll
    for (int r = 0; r < 8; ++r) acc[r] = bv;

    const float* __restrict__ xrow = X + (size_t)(m0 + l16) * DIM;

#pragma unroll
    for (int k = 0; k < DIM; k += 4) {
        const int ka = k + half * 2;
        v2f a;
        a.x = xrow[ka];
        a.y = xrow[ka + 1];
        if (RELU_IN) {
            a.x = fmaxf(a.x, 0.0f);
            a.y = fmaxf(a.y, 0.0f);
        }
        v2f b;
        b.x = W[(size_t)ka * DIM + n];
        b.y = W[(size_t)(ka + 1) * DIM + n];
        // 8 args: (neg_a, A, neg_b, B, c_mod, C, reuse_a, reuse_b)
        acc = __builtin_amdgcn_wmma_f32_16x16x4_f32(
            false, a, false, b, (short)0, acc, false, false);
    }

    float* __restrict__ hrow = H + (size_t)(m0 + half * 8) * DIM + n;
#pragma unroll
    for (int r = 0; r < 8; ++r) hrow[(size_t)r * DIM] = acc[r];
}

// ---------------------------------------------------------------------------
// AGG[col] += dis[row]*dis[col] * H[row]   (edges then self loops)
// One wave per (edge|self-loop); each lane handles 4 of the 128 dims.
// H (51.2 MB) is L2-resident, so gathers + f32 atomic adds stay in L2.
// ---------------------------------------------------------------------------
__global__ void scatter_kernel(const float* __restrict__ H,
                               const float* __restrict__ dis,
                               const int* __restrict__ row,
                               const int* __restrict__ col,
                               float* __restrict__ AGG) {
    const long long gid  = (long long)blockIdx.x * blockDim.x + threadIdx.x;
    const long long item = gid >> 5;
    const int       lane = (int)(gid & 31);
    const long long total = (long long)N_EDGES + N_NODES;
    if (item >= total) return;

    int r, c;
    if (item < N_EDGES) {
        r = row[item];
        c = col[item];
    } else {
        r = c = (int)(item - N_EDGES);   // self loop
    }
    const float nrm = dis[r] * dis[c];
    const int d = lane << 2;

    const float4 h = *(const float4*)(H + (size_t)r * DIM + d);
    float* out = AGG + (size_t)c * DIM + d;
    atomicAdd(out + 0, nrm * h.x);
    atomicAdd(out + 1, nrm * h.y);
    atomicAdd(out + 2, nrm * h.z);
    atomicAdd(out + 3, nrm * h.w);
}

// ---------------------------------------------------------------------------
// Row-wise softmax over DIM=128, one wave per row (32 lanes x float4).
// ---------------------------------------------------------------------------
__global__ void softmax_kernel(float* __restrict__ A) {
    const int wave = threadIdx.x >> 5;
    const int lane = threadIdx.x & 31;
    const int rowi = blockIdx.x * 8 + wave;
    if (rowi >= N_NODES) return;

    float* p = A + (size_t)rowi * DIM + lane * 4;
    float4 v = *(float4*)p;

    float m = fmaxf(fmaxf(v.x, v.y), fmaxf(v.z, v.w));
#pragma unroll
    for (int off = 16; off > 0; off >>= 1) m = fmaxf(m, __shfl_xor(m, off, 32));

    v.x = expf(v.x - m);
    v.y = expf(v.y - m);
    v.z = expf(v.z - m);
    v.w = expf(v.w - m);

    float s = v.x + v.y + v.z + v.w;
#pragma unroll
    for (int off = 16; off > 0; off >>= 1) s += __shfl_xor(s, off, 32);

    const float inv = 1.0f / s;
    v.x *= inv; v.y *= inv; v.z *= inv; v.w *= inv;
    *(float4*)p = v;
}

// ---------------------------------------------------------------------------
// Scores: one wave per batch element.
//   pos_score = ||F[h] + R[e]  - F[t] ||2
//   ng_score  = ||F[h] + R[ne] - F[nt]||2
//   pos_ypre  = sigmoid(sum w0*th*tt + w1*th*tr + w2*tt*tr),  t* = tanh(.)
//   ng_ypre   = same with tail -> ng_tail (relation stays R[e])
// ---------------------------------------------------------------------------
__global__ void score_kernel(const float* __restrict__ FINAL,
                             const float* __restrict__ rel_emb,
                             const int* __restrict__ head,
                             const int* __restrict__ edge,
                             const int* __restrict__ tail,
                             const int* __restrict__ ng_tail,
                             const int* __restrict__ ng_rela,
                             const float* __restrict__ weight_add,
                             float* __restrict__ out) {
    const int wave = threadIdx.x >> 5;
    const int lane = threadIdx.x & 31;
    const int i = blockIdx.x * 8 + wave;
    if (i >= BATCH) return;
    const int d = lane * 4;

    const float w0 = weight_add[0];
    const float w1 = weight_add[1];
    const float w2 = weight_add[2];

    const float4 h4  = *(const float4*)(FINAL  + (size_t)head[i]   * DIM + d);
    const float4 t4  = *(const float4*)(FINAL  + (size_t)tail[i]   * DIM + d);
    const float4 n4  = *(const float4*)(FINAL  + (size_t)ng_tail[i]* DIM + d);
    const float4 r4  = *(const float4*)(rel_emb + (size_t)edge[i]   * DIM + d);
    const float4 q4  = *(const float4*)(rel_emb + (size_t)ng_rela[i]* DIM + d);

    const float ha[4] = {h4.x, h4.y, h4.z, h4.w};
    const float ta[4] = {t4.x, t4.y, t4.z, t4.w};
    const float na[4] = {n4.x, n4.y, n4.z, n4.w};
    const float ra[4] = {r4.x, r4.y, r4.z, r4.w};
    const float qa[4] = {q4.x, q4.y, q4.z, q4.w};

    float ps = 0.0f, ns = 0.0f, py = 0.0f, ny = 0.0f;
#pragma unroll
    for (int j = 0; j < 4; ++j) {
        const float dp = ha[j] + ra[j] - ta[j];
        ps += dp * dp;
        const float dn = ha[j] + qa[j] - na[j];
        ns += dn * dn;
        const float th  = tanhf(ha[j]);
        const float tt  = tanhf(ta[j]);
        const float tnt = tanhf(na[j]);
        const float tr  = tanhf(ra[j]);
        py += w0 * th * tt  + w1 * th * tr + w2 * tt  * tr;
        ny += w0 * th * tnt + w1 * th * tr + w2 * tnt * tr;
    }
#pragma unroll
    for (int off = 16; off > 0; off >>= 1) {
        ps += __shfl_xor(ps, off, 32);
        ns += __shfl_xor(ns, off, 32);
        py += __shfl_xor(py, off, 32);
        ny += __shfl_xor(ny, off, 32);
    }
    if (lane == 0) {
        out[i]             = sqrtf(ps);
        out[BATCH + i]     = sqrtf(ns);
        out[2 * BATCH + i] = 1.0f / (1.0f + expf(-py));
        out[3 * BATCH + i] = 1.0f / (1.0f + expf(-ny));
    }
}

// ---------------------------------------------------------------------------
// Orchestration. Workspace layout (floats): dis | H | AGG  (~103 MB total)
// ---------------------------------------------------------------------------
extern "C" void kernel_launch(void* const* d_in, const int* in_sizes, int n_in,
                              void* d_out, int out_size, void* d_ws, size_t ws_size,
                              hipStream_t stream) {
    const int*   edge_index = (const int*)d_in[0];
    const int*   head       = (const int*)d_in[1];
    const int*   edge       = (const int*)d_in[2];
    const int*   tail       = (const int*)d_in[3];
    const int*   ng_tail    = (const int*)d_in[4];
    const int*   ng_rela    = (const int*)d_in[5];
    const float* node_emb   = (const float*)d_in[6];
    const float* rel_emb    = (const float*)d_in[7];
    const float* W1         = (const float*)d_in[8];
    const float* b1         = (const float*)d_in[9];
    const float* W2         = (const float*)d_in[10];
    const float* b2         = (const float*)d_in[11];
    const float* weight_add = (const float*)d_in[12];

    const int* row = edge_index;            // edge_index[0]
    const int* col = edge_index + N_EDGES;  // edge_index[1]

    float* dis = (float*)d_ws;
    float* H   = dis + ((N_NODES + 255) & ~255);
    float* AGG = H + (size_t)N_NODES * DIM;
    const size_t feat_bytes = (size_t)N_NODES * DIM * sizeof(float);

    // --- degree normalization ---
    init_deg_kernel<<<(N_NODES + 255) / 256, 256, 0, stream>>>(dis);
    deg_accum_kernel<<<(N_EDGES + 255) / 256, 256, 0, stream>>>(row, dis);
    deg_rsqrt_kernel<<<(N_NODES + 255) / 256, 256, 0, stream>>>(dis);

    const long long sc_threads = ((long long)N_EDGES + N_NODES) * 32;
    const int sc_blocks = (int)((sc_threads + 255) / 256);

    // --- conv1: H = X @ W1 + b1 ; AGG = scatter(norm * H[row]) ---
    gemm_wmma_kernel<false><<<N_NODES / 16, 256, 0, stream>>>(node_emb, W1, b1, H);
    hipMemsetAsync(AGG, 0, feat_bytes, stream);
    scatter_kernel<<<sc_blocks, 256, 0, stream>>>(H, dis, row, col, AGG);

    // --- conv2: H = relu(AGG) @ W2 + b2 ; AGG = scatter(norm * H[row]) ---
    gemm_wmma_kernel<true><<<N_NODES / 16, 256, 0, stream>>>(AGG, W2, b2, H);
    hipMemsetAsync(AGG, 0, feat_bytes, stream);
    scatter_kernel<<<sc_blocks, 256, 0, stream>>>(H, dis, row, col, AGG);

    // --- softmax in place -> final node embeddings ---
    softmax_kernel<<<N_NODES / 8, 256, 0, stream>>>(AGG);

    // --- batched scores ---
    score_kernel<<<BATCH / 8, 256, 0, stream>>>(AGG, rel_emb, head, edge, tail,
                                                ng_tail, ng_rela, weight_add,
                                                (float*)d_out);
}